// AAGCLayer_56779467653361
// MI455X (gfx1250) — compile-verified
//
#include <hip/hip_runtime.h>
#include <math.h>

typedef __attribute__((ext_vector_type(2))) float v2f;
typedef __attribute__((ext_vector_type(8))) float v8f;

#define BLOCK 256
#define WAVES_PER_BLOCK (BLOCK / 32)
#define NBLOCKS 1024

// sigmoid with hardware rcp (v_rcp_f32) instead of IEEE division:
// v_mul + v_exp_f32 + v_add + v_rcp_f32  (4 VALU, 2 of them TRANS ops)
__device__ __forceinline__ float sigmoid_fast(float x) {
    return __builtin_amdgcn_rcpf(1.0f + __expf(-x));
}

// out[b] = sigmoid(M @ x_b), M from the tiny matrix chain.
// Per wave: 16 samples = 144 columns = 9 WMMA column-tiles of V_WMMA_F32_16X16X4_F32
// (4 chained K-steps -> K=16; row/col 15 zero-padded).
__global__ __launch_bounds__(BLOCK) void aagc_wmma_kernel(
    const float* __restrict__ x,        // [B,135]
    const float* __restrict__ A_init,   // [15,15]
    const float* __restrict__ A_change, // [15,15]
    const float* __restrict__ Hidden,   // [15,50]
    const float* __restrict__ sigma,    // [50,50]
    const float* __restrict__ kernelW,  // [50,15]
    const float* __restrict__ bias,     // [15,50]
    float* __restrict__ out,            // [B,135]
    int nSamples)
{
    __shared__ float sA[225];   // A_cur
    __shared__ float sS[750];   // support
    __shared__ float sH[750];   // Hidden_new
    __shared__ float sMap[225]; // mapfuc
    __shared__ float sM[256];   // M, zero-padded to 16x16

    const int tid = threadIdx.x;

    // ---------- prologue: tiny matrix chain (redundant per block, ~127 KFLOP) ----------
    if (tid < 225) sA[tid] = A_init[tid] + A_change[tid];
    sM[tid] = 0.0f;
    __syncthreads();

    for (int i = tid; i < 750; i += BLOCK) {            // support = sig(A_cur @ Hidden)
        const int r = i / 50, c = i - r * 50;
        float acc = 0.0f;
        #pragma unroll
        for (int k = 0; k < 15; ++k) acc += sA[r * 15 + k] * Hidden[k * 50 + c];
        sS[i] = sigmoid_fast(acc);
    }
    __syncthreads();
    for (int i = tid; i < 750; i += BLOCK) {            // Hn = sig(support @ sigma + bias)
        const int r = i / 50, c = i - r * 50;
        float acc = bias[i];
        for (int k = 0; k < 50; ++k) acc += sS[r * 50 + k] * sigma[k * 50 + c];
        sH[i] = sigmoid_fast(acc);
    }
    __syncthreads();
    if (tid < 225) {                                    // mapfuc = sig(Hn @ kernel)
        const int r = tid / 15, c = tid - r * 15;
        float acc = 0.0f;
        for (int k = 0; k < 50; ++k) acc += sH[r * 50 + k] * kernelW[k * 15 + c];
        sMap[tid] = sigmoid_fast(acc);
    }
    __syncthreads();
    if (tid < 225) {                                    // M = mapfuc @ A_cur (into 16x16 pad)
        const int r = tid / 15, c = tid - r * 15;
        float acc = 0.0f;
        #pragma unroll
        for (int k = 0; k < 15; ++k) acc += sMap[r * 15 + k] * sA[k * 15 + c];
        sM[r * 16 + c] = acc;
    }
    __syncthreads();

    // ---------- wave-resident A-fragments of M (built once, reused for every tile) ----------
    const int lane = tid & 31;
    const int half = lane >> 4;   // 0: lanes 0-15, 1: lanes 16-31
    const int m    = lane & 15;
    v2f afrag[4];
    #pragma unroll
    for (int k = 0; k < 4; ++k) {
        const int col = 4 * k + 2 * half;               // ISA A-layout: x->col, y->col+1
        afrag[k].x = sM[m * 16 + col];
        afrag[k].y = sM[m * 16 + col + 1];
    }

    // ---------- per-lane, tile-invariant column addressing ----------
    // column tile t: local column c = t*16 + m -> sample-in-tile s, feature f
    int cOff[9];   // element offset into the 2160-float tile slab: s*135 + f
    int sIdx[9];   // sample-in-tile (for tail bounds)
    #pragma unroll
    for (int t = 0; t < 9; ++t) {
        const int c = t * 16 + m;
        const int s = c / 9;
        const int f = c - s * 9;
        sIdx[t] = s;
        cOff[t] = s * 135 + f;
    }
    // clamped address for the structurally-padded B element (row 15, k==3, half==1)
    const int offY3 = half ? 0 : 13 * 9;                // p[offY3] always in-bounds

    // ---------- main loop: one 16-sample (144-column) tile per wave iteration ----------
    const int nTiles     = (nSamples + 15) >> 4;
    const int gw         = blockIdx.x * WAVES_PER_BLOCK + (tid >> 5);
    const int totalWaves = gridDim.x * WAVES_PER_BLOCK;

    for (int tile = gw; tile < nTiles; tile += totalWaves) {
        const long long sampBase = (long long)tile * 16;
        const float* __restrict__ slab  = x   + (size_t)tile * 2160u;
        float*       __restrict__ oslab = out + (size_t)tile * 2160u;

        if (sampBase + 16 <= (long long)nSamples) {
            // ======== fast path: full tile, zero guards on loads ========
            // 1) batch ALL 72 B-fragment loads (max memory-level parallelism)
            v2f bf[9][4];
            #pragma unroll
            for (int t = 0; t < 9; ++t) {
                const float* __restrict__ p = slab + cOff[t];
                #pragma unroll
                for (int k = 0; k < 3; ++k) {
                    const int j0 = 4 * k + 2 * half;    // rows j0, j0+1 (<= 14, always valid)
                    bf[t][k].x = p[j0 * 9];
                    bf[t][k].y = p[j0 * 9 + 9];
                }
                // k == 3: rows 12+2*half, 13+2*half; row 15 is padding -> clamp addr + select
                bf[t][3].x = p[(12 + 2 * half) * 9];
                const float vy = p[offY3];
                bf[t][3].y = half ? 0.0f : vy;
            }
            // 2) 36 WMMAs + sigmoid + stores
            #pragma unroll
            for (int t = 0; t < 9; ++t) {
                v8f acc = {};
                #pragma unroll
                for (int k = 0; k < 4; ++k)
                    acc = __builtin_amdgcn_wmma_f32_16x16x4_f32(
                            false, afrag[k], false, bf[t][k], (short)0, acc, false, false);
                float* __restrict__ q = oslab + cOff[t];
                #pragma unroll
                for (int r = 0; r < 7; ++r)             // rows r+8*half, r<7 always valid
                    q[(r + 8 * half) * 9] = sigmoid_fast(acc[r]);
                if (half == 0)                          // r==7: row 7 valid, row 15 is padding
                    q[7 * 9] = sigmoid_fast(acc[7]);
            }
        } else {
            // ======== tail path (at most one partial tile in the whole grid) ========
            #pragma unroll
            for (int t = 0; t < 9; ++t) {
                const long long S   = sampBase + sIdx[t];
                const bool      inb = S < (long long)nSamples;
                const long long Sc  = inb ? S : (long long)(nSamples - 1);
                const float* __restrict__ p = x + (size_t)Sc * 135u + (cOff[t] - sIdx[t] * 135);

                v8f acc = {};
                #pragma unroll
                for (int k = 0; k < 4; ++k) {
                    const int j0 = 4 * k + 2 * half;
                    const int j1 = j0 + 1;
                    v2f b;
                    const float vx = p[j0 * 9];         // j0 <= 14, address clamped in-bounds
                    b.x = inb ? vx : 0.0f;
                    const int offy = (j1 < 15) ? j1 * 9 : 0;
                    const float vy = p[offy];
                    b.y = (inb && j1 < 15) ? vy : 0.0f;
                    acc = __builtin_amdgcn_wmma_f32_16x16x4_f32(
                            false, afrag[k], false, b, (short)0, acc, false, false);
                }
                float* __restrict__ q = out + (size_t)S * 135u + (cOff[t] - sIdx[t] * 135);
                #pragma unroll
                for (int r = 0; r < 8; ++r) {
                    const int i = r + 8 * half;
                    if (inb && i < 15) q[i * 9] = sigmoid_fast(acc[r]);
                }
            }
        }
    }
}

extern "C" void kernel_launch(void* const* d_in, const int* in_sizes, int n_in,
                              void* d_out, int out_size, void* d_ws, size_t ws_size,
                              hipStream_t stream) {
    const float* x        = (const float*)d_in[0];
    const float* A_init   = (const float*)d_in[1];
    const float* A_change = (const float*)d_in[2];
    const float* Hidden   = (const float*)d_in[3];
    const float* sigma    = (const float*)d_in[4];
    const float* kernelW  = (const float*)d_in[5];
    const float* bias     = (const float*)d_in[6];
    float* out = (float*)d_out;

    const int nSamples = in_sizes[0] / 135;             // 524288
    const int nTiles   = (nSamples + 15) / 16;          // 32768
    int blocks = (nTiles + WAVES_PER_BLOCK - 1) / WAVES_PER_BLOCK;
    if (blocks > NBLOCKS) blocks = NBLOCKS;             // 1024 blocks -> 8192 waves -> 4 tiles/wave

    aagc_wmma_kernel<<<blocks, BLOCK, 0, stream>>>(
        x, A_init, A_change, Hidden, sigma, kernelW, bias, out, nSamples);
}